// KVmemNN_58239756533983
// MI455X (gfx1250) — compile-verified
//
#include <hip/hip_runtime.h>
#include <hip/hip_bf16.h>
#include <math.h>

#define D 256

typedef __attribute__((ext_vector_type(2))) float v2f;
typedef __attribute__((ext_vector_type(8))) float v8f;

// ---------------------------------------------------------------------------
// Embedding-bag mean: out[row][d] = mean_j emb[ids[row][j]][d]
// One block per row; 64 threads x float4 = 256 dims. Each id fetch is one
// global_load_b128 per thread (512B per wave) -> max bytes/instruction for
// the HBM-bound gather stage (~550 MB total @ 23.3 TB/s).
// ---------------------------------------------------------------------------
__global__ __launch_bounds__(64) void bag_mean_kernel(
    const int* __restrict__ ids, int L,
    const float* __restrict__ emb, float* __restrict__ out) {
  const int row = blockIdx.x;
  const int t = threadIdx.x;                 // 0..63, owns dims [4t, 4t+3]
  const int* rids = ids + (size_t)row * L;
  float4 acc = make_float4(0.f, 0.f, 0.f, 0.f);
  for (int j = 0; j < L; ++j) {
    const float4 e = ((const float4*)(emb + (size_t)rids[j] * D))[t];
    acc.x += e.x; acc.y += e.y; acc.z += e.z; acc.w += e.w;
  }
  const float inv = 1.0f / (float)L;
  acc.x *= inv; acc.y *= inv; acc.z *= inv; acc.w *= inv;
  ((float4*)(out + (size_t)row * D))[t] = acc;
}

// ---------------------------------------------------------------------------
// Persona attention hop: cosine(enc_q, enc_persona[p]) -> softmax(8) ->
// a[d] = sum_p soft[p] * enc_persona[p][d].  Single block of 256 threads.
// ---------------------------------------------------------------------------
__global__ __launch_bounds__(256) void persona_attn_kernel(
    const float* __restrict__ enc_q,   // [256]
    const float* __restrict__ enc_p,   // [8, 256]
    float* __restrict__ a_out) {       // [256]
  __shared__ float red[D];
  __shared__ float sims[8];
  __shared__ float soft[8];
  __shared__ float qn_sh;
  const int d = threadIdx.x;
  const float qd = enc_q[d];

  red[d] = qd * qd;
  __syncthreads();
  for (int s = 128; s > 0; s >>= 1) { if (d < s) red[d] += red[d + s]; __syncthreads(); }
  if (d == 0) qn_sh = fmaxf(sqrtf(red[0]), 1e-6f);
  __syncthreads();
  const float qn = qn_sh;
  __syncthreads();

  for (int p = 0; p < 8; ++p) {
    const float pd = enc_p[p * D + d];
    red[d] = qd * pd;
    __syncthreads();
    for (int s = 128; s > 0; s >>= 1) { if (d < s) red[d] += red[d + s]; __syncthreads(); }
    const float dot = red[0];
    __syncthreads();
    red[d] = pd * pd;
    __syncthreads();
    for (int s = 128; s > 0; s >>= 1) { if (d < s) red[d] += red[d + s]; __syncthreads(); }
    if (d == 0) sims[p] = dot / (qn * fmaxf(sqrtf(red[0]), 1e-6f));
    __syncthreads();
  }

  if (d == 0) {
    float mx = sims[0];
    for (int p = 1; p < 8; ++p) mx = fmaxf(mx, sims[p]);
    float s = 0.f;
    for (int p = 0; p < 8; ++p) { soft[p] = expf(sims[p] - mx); s += soft[p]; }
    const float inv = 1.0f / s;
    for (int p = 0; p < 8; ++p) soft[p] *= inv;
  }
  __syncthreads();

  float acc = 0.f;
  for (int p = 0; p < 8; ++p) acc += soft[p] * enc_p[p * D + d];
  a_out[d] = acc;
}

// ---------------------------------------------------------------------------
// y[i] = sum_j R[i][j] * x[j]   (== (x @ R^T) for a row vector x)
// ---------------------------------------------------------------------------
__global__ __launch_bounds__(256) void rw_matvec_kernel(
    const float* __restrict__ R, const float* __restrict__ x,
    float* __restrict__ y) {
  __shared__ float xs[D];
  const int i = threadIdx.x;
  xs[i] = x[i];
  __syncthreads();
  const float* row = R + (size_t)i * D;
  float acc = 0.f;
  #pragma unroll 8
  for (int j = 0; j < D; ++j) acc += row[j] * xs[j];
  y[i] = acc;
}

// ---------------------------------------------------------------------------
// WMMA matvec: out[m] = dot(A[m][:256], q)  using V_WMMA_F32_16X16X4_F32.
// One wave per 16-row tile; K-loop of 64 chained f32 WMMAs.
// A layout (ISA 7.12.2, 32-bit A 16x4): lanes 0-15 = M rows,
//   VGPR0 = {K0 | K2 (hi lanes)}, VGPR1 = {K1 | K3 (hi lanes)}.
// B is q[k] broadcast across all 16 N columns (robust to N-lane mapping),
// so every column of D equals the score vector; lanes 0/16 write it out
// per the documented C/D striping (VGPR r -> M=r / M=r+8).
// ---------------------------------------------------------------------------
__global__ __launch_bounds__(256) void matvec_wmma_kernel(
    const float* __restrict__ A,   // [M, 256]
    const float* __restrict__ q,   // [256]
    float* __restrict__ out, int M) {
  const int lane = threadIdx.x & 31;
  const int wave = threadIdx.x >> 5;
  const int tile = blockIdx.x * (blockDim.x >> 5) + wave;
  const int m0 = tile * 16;
  if (m0 >= M) return;  // uniform per wave; active waves keep EXEC all-1s

  const int half = lane >> 4;   // 0: K0/K1, 1: K2/K3
  const int r = lane & 15;      // A-matrix row within tile
  const float* __restrict__ arow = A + (size_t)(m0 + r) * D;

  v8f c = {};
  #pragma unroll 8
  for (int k0 = 0; k0 < D; k0 += 4) {
    const int ka = k0 + half * 2;
    v2f a; a.x = arow[ka];  a.y = arow[ka + 1];
    v2f b; b.x = q[ka];     b.y = q[ka + 1];
    c = __builtin_amdgcn_wmma_f32_16x16x4_f32(
        /*neg_a=*/false, a, /*neg_b=*/false, b,
        /*c_mod=*/(short)0, c, /*reuse_a=*/false, /*reuse_b=*/false);
  }

  if (lane == 0) {
    #pragma unroll
    for (int i = 0; i < 8; ++i) out[m0 + i] = c[i];
  } else if (lane == 16) {
    #pragma unroll
    for (int i = 0; i < 8; ++i) out[m0 + 8 + i] = c[i];
  }
}

// ---------------------------------------------------------------------------
// Softmax over N elements (single block, 1024 threads).
// ---------------------------------------------------------------------------
__global__ __launch_bounds__(1024) void softmax_kernel(
    const float* __restrict__ in, float* __restrict__ out, int N) {
  __shared__ float red[1024];
  const int t = threadIdx.x;
  float mx = -3.402823466e+38f;
  for (int i = t; i < N; i += 1024) mx = fmaxf(mx, in[i]);
  red[t] = mx;
  __syncthreads();
  for (int s = 512; s > 0; s >>= 1) { if (t < s) red[t] = fmaxf(red[t], red[t + s]); __syncthreads(); }
  mx = red[0];
  __syncthreads();
  float sum = 0.f;
  for (int i = t; i < N; i += 1024) sum += expf(in[i] - mx);
  red[t] = sum;
  __syncthreads();
  for (int s = 512; s > 0; s >>= 1) { if (t < s) red[t] += red[t + s]; __syncthreads(); }
  const float inv = 1.0f / red[0];
  for (int i = t; i < N; i += 1024) out[i] = expf(in[i] - mx) * inv;
}

// ---------------------------------------------------------------------------
// Probability-weighted sum of enc_values (deterministic two-phase, no float
// atomics).  64 threads x float4 per block; block b covers an M/gridDim row
// chunk: partial[b][d] = sum p[m]*vals[m][d].
// ---------------------------------------------------------------------------
__global__ __launch_bounds__(64) void weighted_sum_partial_kernel(
    const float* __restrict__ probs, const float* __restrict__ vals, // [M,256]
    float* __restrict__ partial, int M) {
  const int t = threadIdx.x;                 // 0..63, owns dims [4t, 4t+3]
  const int rows_per_block = M / gridDim.x;
  const int m0 = blockIdx.x * rows_per_block;
  float4 acc = make_float4(0.f, 0.f, 0.f, 0.f);
  for (int m = m0; m < m0 + rows_per_block; ++m) {
    const float p = probs[m];
    const float4 v = ((const float4*)(vals + (size_t)m * D))[t];
    acc.x += p * v.x; acc.y += p * v.y; acc.z += p * v.z; acc.w += p * v.w;
  }
  ((float4*)(partial + (size_t)blockIdx.x * D))[t] = acc;
}

__global__ __launch_bounds__(256) void reduce_partials_kernel(
    const float* __restrict__ partial, float* __restrict__ out, int nb) {
  const int d = threadIdx.x;
  float acc = 0.f;
  for (int b = 0; b < nb; ++b) acc += partial[b * D + d];
  out[d] = acc;
}

// ---------------------------------------------------------------------------
// Launch
// ---------------------------------------------------------------------------
extern "C" void kernel_launch(void* const* d_in, const int* in_sizes, int n_in,
                              void* d_out, int out_size, void* d_ws, size_t ws_size,
                              hipStream_t stream) {
  const int*   xs     = (const int*)d_in[0];    // [1,64]
  const int*   cand   = (const int*)d_in[1];    // [1024,24]
  const int*   pers   = (const int*)d_in[2];    // [8,16]
  /* d_in[3] = label, unused by the reference */
  const int*   keys   = (const int*)d_in[4];    // [8192,32]
  const int*   values = (const int*)d_in[5];    // [8192,32]
  const float* emb    = (const float*)d_in[6];  // [50000,256]
  const float* Rw     = (const float*)d_in[7];  // [256,256]

  const int M_KV = 8192, C = 1024;

  float* ws = (float*)d_ws;
  float* enc_keys    = ws;                               // 8192*256
  float* enc_values  = enc_keys    + (size_t)M_KV * D;   // 8192*256
  float* enc_cands   = enc_values  + (size_t)M_KV * D;   // 1024*256
  float* enc_persona = enc_cands   + (size_t)C * D;      // 8*256
  float* enc_q       = enc_persona + 8 * D;              // 256
  float* a1          = enc_q       + D;                  // 256
  float* q1          = a1          + D;                  // 256
  float* scores      = q1          + D;                  // 8192
  float* probs       = scores      + M_KV;               // 8192
  float* partials    = probs       + M_KV;               // 64*256
  float* a2          = partials    + 64 * D;             // 256
  float* q2          = a2          + D;                  // 256
  float* logits      = q2          + D;                  // 1024
  // total ~4.5M floats (~18 MB) of workspace

  // 1) Embedding-bag means (the HBM-bound stage: ~550 MB of b128 gather)
  bag_mean_kernel<<<M_KV, 64, 0, stream>>>(keys,   32, emb, enc_keys);
  bag_mean_kernel<<<M_KV, 64, 0, stream>>>(values, 32, emb, enc_values);
  bag_mean_kernel<<<C,    64, 0, stream>>>(cand,   24, emb, enc_cands);
  bag_mean_kernel<<<8,    64, 0, stream>>>(pers,   16, emb, enc_persona);
  bag_mean_kernel<<<1,    64, 0, stream>>>(xs,     64, emb, enc_q);

  // 2) Persona attention hop -> q1 = (soft @ enc_persona) @ Rw^T
  persona_attn_kernel<<<1, 256, 0, stream>>>(enc_q, enc_persona, a1);
  rw_matvec_kernel<<<1, 256, 0, stream>>>(Rw, a1, q1);

  // 3) Key-value memory hop (WMMA scores -> softmax -> weighted values -> Rw^T)
  matvec_wmma_kernel<<<M_KV / 128, 256, 0, stream>>>(enc_keys, q1, scores, M_KV);
  softmax_kernel<<<1, 1024, 0, stream>>>(scores, probs, M_KV);
  weighted_sum_partial_kernel<<<64, 64, 0, stream>>>(probs, enc_values, partials, M_KV);
  reduce_partials_kernel<<<1, 256, 0, stream>>>(partials, a2, 64);
  rw_matvec_kernel<<<1, 256, 0, stream>>>(Rw, a2, q2);

  // 4) Candidate scoring (WMMA) + softmax -> d_out [1024]
  matvec_wmma_kernel<<<C / 128, 256, 0, stream>>>(enc_cands, q2, logits, C);
  softmax_kernel<<<1, 1024, 0, stream>>>(logits, (float*)d_out, C);
}